// SS2D_32521492365679
// MI455X (gfx1250) — compile-verified
//
#include <hip/hip_runtime.h>
#include <hip/hip_bf16.h>
#include <math.h>

// ---------------- problem constants ----------------
#define B_SZ   2
#define D_IN   192
#define H_IMG  48
#define W_IMG  48
#define L_TOT  (H_IMG * W_IMG)      // 2304
#define K_DIR  4
#define N_ST   16
#define R_DT   6
#define C_PROJ (R_DT + 2 * N_ST)    // 38
#define C_PAD  48                   // padded to 3 WMMA M-tiles
#define NCHUNK 16
#define CHLEN  (L_TOT / NCHUNK)     // 144
#define NSEQ   (B_SZ * K_DIR * D_IN) // 1536
#define LT_PER_BLK 8                // l-tiles per projection block (1 per wave)

typedef __attribute__((ext_vector_type(2))) float v2f;
typedef __attribute__((ext_vector_type(8))) float v8f;

// cross-scan index map: xs[b,k,d,l] == x[b,d, xs_map(k,l)]
__device__ __forceinline__ int xs_map(int k, int l) {
  int lf = (k & 2) ? (L_TOT - 1 - l) : l;     // dirs 2,3 are reversed
  if (k & 1) {                                // dirs 1,3 are transposed
    int w = lf / H_IMG;
    int h = lf % H_IMG;
    lf = h * W_IMG + w;
  }
  return lf;
}

__device__ __forceinline__ float softplus_f(float t) {
  return (t > 20.0f) ? t : log1pf(__expf(t));
}

// ---------------------------------------------------------------------------
// Kernel 1: x_dbl[b,k,c,l] = sum_d W[k,c,d] * xs[b,k,d,l]   via fp32 WMMA.
// 256-thread block (8 waves); weights zero-padded to 48x192 staged in LDS
// once per block, so the A operand is an unconditional ds_load_b64 (no
// exec-divergent guards in the WMMA loop). Each wave owns one 16-wide l-tile.
// A layout (32-bit 16x4): lanes 0-15 -> M=lane, v0:K=0 v1:K=1;
//                         lanes 16-31 -> M=lane-16, v0:K=2 v1:K=3.
// B layout mirrors with N in place of M. C/D: vgpr r holds M=r (+8 for hi half).
// ---------------------------------------------------------------------------
__global__ __launch_bounds__(256) void ss2d_proj_wmma(
    const float* __restrict__ x, const float* __restrict__ Wp,
    float* __restrict__ x_dbl) {
  __shared__ float sW[C_PAD * D_IN];          // 36.9 KB of 320 KB/WGP

  int gid = blockIdx.x;
  int lblk = gid % (L_TOT / 16 / LT_PER_BLK); gid /= (L_TOT / 16 / LT_PER_BLK);
  int k = gid % K_DIR;
  int b = gid / K_DIR;

  const float* __restrict__ Wk = Wp + (size_t)k * C_PROJ * D_IN;
  for (int i = threadIdx.x; i < C_PAD * D_IN; i += 256) {
    const int c = i / D_IN, dd = i % D_IN;
    sW[i] = (c < C_PROJ) ? Wk[(size_t)c * D_IN + dd] : 0.0f;
  }
  __syncthreads();

  const int wave  = threadIdx.x >> 5;
  const int lane  = threadIdx.x & 31;
  const int n     = lane & 15;
  const int khalf = lane >> 4;                // 0 or 1
  const int ltile = lblk * LT_PER_BLK + wave;
  const int l     = ltile * 16 + n;
  const int lf    = xs_map(k, l);             // B-operand column address in x

  const float* __restrict__ xb = x + (size_t)b * D_IN * L_TOT;

  v8f acc[3];
  acc[0] = v8f{}; acc[1] = v8f{}; acc[2] = v8f{};

  for (int kk = 0; kk < D_IN; kk += 4) {
    const int d0 = kk + khalf * 2;
    v2f bm;
    bm.x = xb[(size_t)(d0 + 0) * L_TOT + lf];
    bm.y = xb[(size_t)(d0 + 1) * L_TOT + lf];
#pragma unroll
    for (int mt = 0; mt < 3; ++mt) {
      const int m = mt * 16 + n;
      const v2f am = *(const v2f*)&sW[m * D_IN + d0];   // ds_load_b64
      acc[mt] = __builtin_amdgcn_wmma_f32_16x16x4_f32(
          false, am, false, bm, (short)0, acc[mt], false, false);
    }
  }

  float* __restrict__ outk = x_dbl + ((size_t)(b * K_DIR + k) * C_PROJ) * L_TOT;
#pragma unroll
  for (int mt = 0; mt < 3; ++mt) {
#pragma unroll
    for (int r = 0; r < 8; ++r) {
      const int crow = mt * 16 + khalf * 8 + r;
      if (crow < C_PROJ) outk[(size_t)crow * L_TOT + l] = acc[mt][r];
    }
  }
}

// ---------------------------------------------------------------------------
// Kernel 2: delta = softplus(dt_w[k] @ x_dbl[:R] + dt_bias)   (one elem/thread)
// ---------------------------------------------------------------------------
__global__ __launch_bounds__(256) void ss2d_delta(
    const float* __restrict__ x_dbl, const float* __restrict__ dtw,
    const float* __restrict__ dtb, float* __restrict__ delta) {
  size_t idx = (size_t)blockIdx.x * 256 + threadIdx.x;
  int l = idx % L_TOT; size_t t = idx / L_TOT;
  int d = t % D_IN; t /= D_IN;
  int k = t % K_DIR; int b = t / K_DIR;

  const float* __restrict__ xr = x_dbl + ((size_t)(b * K_DIR + k) * C_PROJ) * L_TOT;
  const float* __restrict__ wr = dtw + ((size_t)(k * D_IN + d)) * R_DT;
  float acc = dtb[k * D_IN + d];
#pragma unroll
  for (int r = 0; r < R_DT; ++r) acc = fmaf(wr[r], xr[(size_t)r * L_TOT + l], acc);
  delta[((size_t)(b * K_DIR + k) * D_IN + d) * L_TOT + l] = softplus_f(acc);
}

// ---------------------------------------------------------------------------
// Kernel 3a: scan pass 1 — per-chunk local state (h0 = 0) and delta-sum.
// Block = (b,k,chunk), 192 threads = d. B rows staged through LDS, 16 steps/tile.
// ---------------------------------------------------------------------------
__global__ __launch_bounds__(192) void ss2d_scan_pass1(
    const float* __restrict__ x, const float* __restrict__ x_dbl,
    const float* __restrict__ delta, const float* __restrict__ A_logs,
    float* __restrict__ hloc, float* __restrict__ Ssum) {
  __shared__ float sB[16 * N_ST];

  int gid = blockIdx.x;
  int ch = gid % NCHUNK; gid /= NCHUNK;
  int k  = gid % K_DIR;
  int b  = gid / K_DIR;
  const int d = threadIdx.x;

  const size_t bk  = (size_t)(b * K_DIR + k);
  const size_t seq = bk * D_IN + d;

  float An[N_ST];
#pragma unroll
  for (int nn = 0; nn < N_ST; ++nn)
    An[nn] = -__expf(A_logs[(size_t)(k * D_IN + d) * N_ST + nn]);

  const float* __restrict__ xb   = x     + (size_t)b * D_IN * L_TOT + (size_t)d * L_TOT;
  const float* __restrict__ drow = delta + seq * L_TOT;
  const float* __restrict__ Brow = x_dbl + (bk * C_PROJ + R_DT) * L_TOT;

  float h[N_ST];
#pragma unroll
  for (int nn = 0; nn < N_ST; ++nn) h[nn] = 0.0f;
  float S = 0.0f;

  for (int it = 0; it < CHLEN / 16; ++it) {
    const int t0 = ch * CHLEN + it * 16;
    for (int i = threadIdx.x; i < 16 * N_ST; i += 192) {
      const int st = i >> 4, nn = i & 15;
      sB[i] = Brow[(size_t)nn * L_TOT + (t0 + st)];
    }
    __builtin_prefetch(&drow[t0 + 16], 0, 0);   // global_prefetch_b8
    __syncthreads();
#pragma unroll 4
    for (int st = 0; st < 16; ++st) {
      const int l = t0 + st;
      const float dl = drow[l];
      const float xv = xb[xs_map(k, l)];
      S += dl;
      const float dx = dl * xv;
#pragma unroll
      for (int nn = 0; nn < N_ST; ++nn)
        h[nn] = fmaf(__expf(dl * An[nn]), h[nn], dx * sB[st * N_ST + nn]);
    }
    __syncthreads();
  }

#pragma unroll
  for (int nn = 0; nn < N_ST; ++nn)
    hloc[(seq * NCHUNK + ch) * N_ST + nn] = h[nn];
  Ssum[seq * NCHUNK + ch] = S;
}

// ---------------------------------------------------------------------------
// Kernel 3b: serial combine over the 16 chunks (one thread per sequence).
// Carry: H <- exp(A * Ssum_chunk) * H + h_local_chunk; record entry state.
// ---------------------------------------------------------------------------
__global__ __launch_bounds__(192) void ss2d_scan_combine(
    const float* __restrict__ A_logs, const float* __restrict__ hloc,
    const float* __restrict__ Ssum, float* __restrict__ hin) {
  const int s = blockIdx.x * 192 + threadIdx.x;   // 0..NSEQ-1
  const int d = s % D_IN;
  const int k = (s / D_IN) % K_DIR;

  float An[N_ST], H[N_ST];
#pragma unroll
  for (int nn = 0; nn < N_ST; ++nn) {
    An[nn] = -__expf(A_logs[(size_t)(k * D_IN + d) * N_ST + nn]);
    H[nn] = 0.0f;
  }
  for (int ch = 0; ch < NCHUNK; ++ch) {
    const size_t base = ((size_t)s * NCHUNK + ch) * N_ST;
    const float Sv = Ssum[(size_t)s * NCHUNK + ch];
#pragma unroll
    for (int nn = 0; nn < N_ST; ++nn) {
      hin[base + nn] = H[nn];                        // state entering chunk ch
      H[nn] = fmaf(__expf(An[nn] * Sv), H[nn], hloc[base + nn]);
    }
  }
}

// ---------------------------------------------------------------------------
// Kernel 3c: scan pass 2 — replay chunks from correct entry state, emit
// out_y = <h, C> + Ds * xs.  out_y may alias delta (same-thread read->write).
// ---------------------------------------------------------------------------
__global__ __launch_bounds__(192) void ss2d_scan_pass2(
    const float* __restrict__ x, const float* __restrict__ x_dbl,
    const float* __restrict__ delta, const float* __restrict__ A_logs,
    const float* __restrict__ Ds, const float* __restrict__ hin,
    float* __restrict__ out_y) {
  __shared__ float sB[16 * N_ST];
  __shared__ float sC[16 * N_ST];

  int gid = blockIdx.x;
  int ch = gid % NCHUNK; gid /= NCHUNK;
  int k  = gid % K_DIR;
  int b  = gid / K_DIR;
  const int d = threadIdx.x;

  const size_t bk  = (size_t)(b * K_DIR + k);
  const size_t seq = bk * D_IN + d;

  float An[N_ST], h[N_ST];
#pragma unroll
  for (int nn = 0; nn < N_ST; ++nn) {
    An[nn] = -__expf(A_logs[(size_t)(k * D_IN + d) * N_ST + nn]);
    h[nn] = hin[(seq * NCHUNK + ch) * N_ST + nn];
  }
  const float dskip = Ds[k * D_IN + d];

  const float* __restrict__ xb   = x     + (size_t)b * D_IN * L_TOT + (size_t)d * L_TOT;
  const float* __restrict__ drow = delta + seq * L_TOT;
  const float* __restrict__ Brow = x_dbl + (bk * C_PROJ + R_DT) * L_TOT;
  const float* __restrict__ Crow = x_dbl + (bk * C_PROJ + R_DT + N_ST) * L_TOT;
  float* __restrict__ yrow = out_y + seq * L_TOT;

  for (int it = 0; it < CHLEN / 16; ++it) {
    const int t0 = ch * CHLEN + it * 16;
    for (int i = threadIdx.x; i < 16 * N_ST; i += 192) {
      const int st = i >> 4, nn = i & 15;
      sB[i] = Brow[(size_t)nn * L_TOT + (t0 + st)];
      sC[i] = Crow[(size_t)nn * L_TOT + (t0 + st)];
    }
    __builtin_prefetch(&drow[t0 + 16], 0, 0);
    __syncthreads();
#pragma unroll 4
    for (int st = 0; st < 16; ++st) {
      const int l = t0 + st;
      const float dl = drow[l];
      const float xv = xb[xs_map(k, l)];
      const float dx = dl * xv;
      float y = 0.0f;
#pragma unroll
      for (int nn = 0; nn < N_ST; ++nn) {
        h[nn] = fmaf(__expf(dl * An[nn]), h[nn], dx * sB[st * N_ST + nn]);
        y = fmaf(h[nn], sC[st * N_ST + nn], y);
      }
      yrow[l] = fmaf(dskip, xv, y);   // safe even if out_y aliases delta
    }
    __syncthreads();
  }
}

// ---------------------------------------------------------------------------
// Kernel 4: merge 4 directions (un-flip / un-transpose) + LayerNorm over D.
// Block = (b,l) with 192 threads (= D). wave32 shuffle + LDS reduction.
// Output layout (B,H,W,D).
// ---------------------------------------------------------------------------
__global__ __launch_bounds__(192) void ss2d_merge_ln(
    const float* __restrict__ out_y, const float* __restrict__ lnw,
    const float* __restrict__ lnb, float* __restrict__ out) {
  __shared__ float s_sum[8];
  __shared__ float s_sq[8];

  const int b = blockIdx.x / L_TOT;
  const int l = blockIdx.x % L_TOT;
  const int d = threadIdx.x;
  const int h = l / W_IMG, w = l % W_IMG;
  const int lt = w * H_IMG + h;           // transposed sequence position

  const size_t bd = ((size_t)b * K_DIR) * D_IN + d;   // dir stride = D_IN rows
  #define OY(kk, ll) out_y[((bd + (size_t)(kk) * D_IN)) * L_TOT + (ll)]
  float y = OY(0, l) + OY(2, L_TOT - 1 - l) + OY(1, lt) + OY(3, L_TOT - 1 - lt);
  #undef OY

  // block reduction for mean / var (6 waves of 32)
  float v = y, v2 = y * y;
#pragma unroll
  for (int off = 16; off > 0; off >>= 1) {
    v  += __shfl_down(v, off);
    v2 += __shfl_down(v2, off);
  }
  const int wid = threadIdx.x >> 5;
  if ((threadIdx.x & 31) == 0) { s_sum[wid] = v; s_sq[wid] = v2; }
  __syncthreads();
  if (threadIdx.x == 0) {
    float a = 0.0f, q = 0.0f;
#pragma unroll
    for (int i = 0; i < 6; ++i) { a += s_sum[i]; q += s_sq[i]; }
    s_sum[6] = a; s_sq[6] = q;
  }
  __syncthreads();
  const float mu  = s_sum[6] * (1.0f / D_IN);
  const float var = s_sq[6] * (1.0f / D_IN) - mu * mu;
  const float r   = rsqrtf(var + 1e-5f);

  out[(((size_t)b * H_IMG + h) * W_IMG + w) * D_IN + d] =
      (y - mu) * r * lnw[d] + lnb[d];
}

// ---------------------------------------------------------------------------
extern "C" void kernel_launch(void* const* d_in, const int* in_sizes, int n_in,
                              void* d_out, int out_size, void* d_ws, size_t ws_size,
                              hipStream_t stream) {
  const float* x    = (const float*)d_in[0];
  const float* xpw  = (const float*)d_in[1];
  const float* dtw  = (const float*)d_in[2];
  const float* dtb  = (const float*)d_in[3];
  const float* Alog = (const float*)d_in[4];
  const float* Ds   = (const float*)d_in[5];
  const float* lnw  = (const float*)d_in[6];
  const float* lnb  = (const float*)d_in[7];
  float* out = (float*)d_out;

  // workspace layout (floats), ~19.3 MB total
  float* ws = (float*)d_ws;
  const size_t n_xdbl  = (size_t)B_SZ * K_DIR * C_PROJ * L_TOT;   // 700,416
  const size_t n_delta = (size_t)B_SZ * K_DIR * D_IN * L_TOT;     // 3,538,944
  const size_t n_hloc  = (size_t)NSEQ * NCHUNK * N_ST;            // 393,216
  const size_t n_S     = (size_t)NSEQ * NCHUNK;                   //  24,576
  float* x_dbl = ws;
  float* delta = x_dbl + n_xdbl;
  float* hloc  = delta + n_delta;
  float* Ssum  = hloc + n_hloc;
  float* hin   = Ssum + n_S;
  float* out_y = delta;   // alias: pass 2 reads delta[l] then writes out_y[l]
  (void)n_in; (void)in_sizes; (void)out_size; (void)ws_size;

  ss2d_proj_wmma<<<B_SZ * K_DIR * (L_TOT / 16 / LT_PER_BLK), 256, 0, stream>>>(
      x, xpw, x_dbl);
  ss2d_delta<<<(B_SZ * K_DIR * D_IN * L_TOT) / 256, 256, 0, stream>>>(
      x_dbl, dtw, dtb, delta);
  ss2d_scan_pass1<<<B_SZ * K_DIR * NCHUNK, 192, 0, stream>>>(
      x, x_dbl, delta, Alog, hloc, Ssum);
  ss2d_scan_combine<<<NSEQ / 192, 192, 0, stream>>>(Alog, hloc, Ssum, hin);
  ss2d_scan_pass2<<<B_SZ * K_DIR * NCHUNK, 192, 0, stream>>>(
      x, x_dbl, delta, Alog, Ds, hin, out_y);
  ss2d_merge_ln<<<B_SZ * L_TOT, 192, 0, stream>>>(out_y, lnw, lnb, out);
}